// GeneralizedCTC_41686952575661
// MI455X (gfx1250) — compile-verified
//
#include <hip/hip_runtime.h>
#include <stdint.h>
#include <stddef.h>

#define NINF (-__builtin_inff())
#define PINF (__builtin_inff())

namespace {
constexpr int kB = 32;
constexpr int kS = 512;
constexpr int kT = 2048;
constexpr int kRows = 512;                    // computed padded rows 1..512 (orig rows 0..511)
constexpr int kNT   = 256;                    // threads per block = 8 waves
constexpr int kR    = 2;                      // rows per thread
constexpr int kTT   = 8;                      // time columns per LDS tile
constexpr int kNBUF = 3;                      // tile ring buffers (prefetch depth 2)
constexpr int kNTiles = kT / kTT;             // 256
constexpr int kTileFloats = kNBUF * kTT * kRows;       // 12288 floats (48 KB)
constexpr int kSmemFloats = kTileFloats + 2 * kNT;     // + boundary double buffer
constexpr int kIssues = (kRows * kTT) / kNT;  // 16 async b32 issues / thread / stage
static_assert(kR * kNT == kRows, "row coverage");
static_assert((kRows * kTT) % kNT == 0, "stage coverage");
static_assert(3 * kIssues <= 63, "ASYNCcnt is 6 bits; never exceed 63 outstanding");
}

// ---- CDNA5 async global->LDS (gfx1250) --------------------------------------
__device__ __forceinline__ uint32_t lds_addr_u32(const void* p) {
  // Low 32 bits of a flat pointer into the LDS aperture are the wave-relative
  // LDS byte address (ISA 10.2 aperture mapping).
  return (uint32_t)(uintptr_t)p;
}
__device__ __forceinline__ void async_ld_b32(uint32_t lds_byte, const float* gsrc) {
  asm volatile("global_load_async_to_lds_b32 %0, %1, off"
               :: "v"(lds_byte), "v"(gsrc) : "memory");
}
// ASYNCcnt <= 16 : oldest of two outstanding stages retired (steady state)
__device__ __forceinline__ void wait_async_steady() { asm volatile("s_wait_asynccnt 16" ::: "memory"); }
// ASYNCcnt <= 32 : stage 0 of the three prologue stages retired
__device__ __forceinline__ void wait_async_prolog() { asm volatile("s_wait_asynccnt 32" ::: "memory"); }
// full drain (needed for the final tile, which has no successor in flight)
__device__ __forceinline__ void wait_async_zero()   { asm volatile("s_wait_asynccnt 0"  ::: "memory"); }

// ---- stable logaddexp -------------------------------------------------------
__device__ __forceinline__ float lae(float x, float y) {
  float m = fmaxf(x, y);
  float d = fminf(x, y) - m;             // d <= 0 ; NaN only if both -inf
  float t = __logf(1.0f + __expf(d));    // exp(-inf)=0 -> t=0
  return (m == NINF) ? NINF : (m + t);
}

// flip map: computed padded row r = j+1; region iff (j < l1 && c < l2)
__device__ __forceinline__ void flip_pos(int j, int c, int l1, int l2,
                                         int& orow, int& ocol) {
  bool inreg = (j < l1) && (c < l2);
  orow = inreg ? (l1 - 1 - j) : j;
  ocol = inreg ? (l2 - 1 - c) : c;
}

// stage one tile (kTT columns x 512 rows) into ring buffer `buf` via async DMA
template <int MODE>   // 0 = fwd (identity gather), 1/2 = bwd (flip gather)
__device__ __forceinline__ void stage_tile(uint32_t smem_base, const float* lpb,
                                           int buf, int c0, int l1, int l2, int tid) {
  uint32_t lds0 = smem_base + (uint32_t)(buf * kTT * kRows) * 4u;
#pragma unroll
  for (int i = 0; i < kIssues; ++i) {
    int e  = tid + i * kNT;
    int tt = e % kTT;
    int j  = e / kTT;
    int c  = c0 + tt;
    const float* src;
    if (MODE == 0) {
      src = lpb + (size_t)j * kT + c;
    } else {
      int grow, gcol;
      flip_pos(j, c, l1, l2, grow, gcol);
      src = lpb + (size_t)grow * kT + gcol;   // backward0[r][c] == lp[flip(r,c)]
    }
    async_ld_b32(lds0 + (uint32_t)(tt * kRows + j) * 4u, src);
  }
}

template <int MODE>   // 0: fwd -> dst   1: bwd raw -> dst(ws)   2: bwd combine in dst(out)
__device__ __forceinline__ void emit(float* dstb, int j, int c, float aval,
                                     float inval, int l1, int l2) {
  if (MODE == 0) {
    dstb[(size_t)j * kT + c] = aval;
  } else {
    int orow, ocol;
    flip_pos(j, c, l1, l2, orow, ocol);
    size_t pos = (size_t)orow * kT + ocol;
    if (MODE == 1) {
      dstb[pos] = aval;
    } else {
      float v = dstb[pos] + aval - inval;     // fwd + bwd - lp  (inval == lp[pos])
      if (v != v || v == NINF) v = NINF;
      else if (v == PINF)      v = 3.4028234663852886e38f;
      dstb[pos] = v;
    }
  }
}

template <int MODE>
__global__ void __launch_bounds__(kNT)
dp_kernel(const float* __restrict__ lp, const int* __restrict__ tls,
          const int* __restrict__ ils, float* __restrict__ dst) {
  extern __shared__ float smem[];
  const int b   = blockIdx.x;
  const int tid = threadIdx.x;
  const int l1  = (MODE == 0) ? 0 : tls[b];
  const int l2  = (MODE == 0) ? 0 : ils[b];
  const float* lpb = lp + (size_t)b * kS * kT;
  float* dstb = dst + (size_t)b * kS * kT;
  const uint32_t smem_base = lds_addr_u32(smem);
  float* bnd = smem + kTileFloats;            // [2][kNT] boundary double buffer
  const int r0 = kR * tid + 1;                // first owned padded row

  // prologue: stage tiles 0,1,2 (48 outstanding <= 63); wait for tile 0
  stage_tile<MODE>(smem_base, lpb, 0, 0 * kTT, l1, l2, tid);
  stage_tile<MODE>(smem_base, lpb, 1, 1 * kTT, l1, l2, tid);
  stage_tile<MODE>(smem_base, lpb, 2, 2 * kTT, l1, l2, tid);
  wait_async_prolog();
  __syncthreads();

  // t = 0 : a0 = col0 with rows!=1 set to -inf
  float a[kR];
#pragma unroll
  for (int jj = 0; jj < kR; ++jj) {
    const int j = r0 - 1 + jj;
    float inval = smem[(0 * kTT + 0) * kRows + j];
    a[jj] = (r0 + jj == 1) ? inval : NINF;
    emit<MODE>(dstb, j, 0, a[jj], inval, l1, l2);
  }
  int pb = 0;
  bnd[pb * kNT + tid] = a[kR - 1];
  __syncthreads();

  for (int t = 1; t < kT; ++t) {
    const int tt  = t % kTT;
    const int k   = t / kTT;
    const int buf = k % kNBUF;
    if (tt == 0) {
      if (k == kNTiles - 1) wait_async_zero();   // no successor stage in flight
      else                  wait_async_steady(); // tile k fully in LDS (this wave)
      __syncthreads();                           // ...and every other wave's share
      if (k + 2 < kNTiles)
        stage_tile<MODE>(smem_base, lpb, (k + 2) % kNBUF, (k + 2) * kTT, l1, l2, tid);
      if (MODE == 2 && k + 1 < kNTiles) {
        // warm the out lines this thread will read-modify-write in the next tile
        const int cn = (k + 1) * kTT;
#pragma unroll
        for (int jj = 0; jj < kR; ++jj) {
          int orow, ocol;
          flip_pos(r0 - 1 + jj, cn, l1, l2, orow, ocol);
          __builtin_prefetch(&dstb[(size_t)orow * kT + ocol], 0, 3);
        }
      }
    }
    float carry = (tid == 0) ? NINF : bnd[pb * kNT + (tid - 1)];  // alpha[t-1][r0-1]
#pragma unroll
    for (int jj = 0; jj < kR; ++jj) {
      const int j = r0 - 1 + jj;
      float inval = smem[(buf * kTT + tt) * kRows + j];
      float old   = a[jj];
      float nv    = inval + lae(carry, old);
      if (tid == 0 && jj == 0) nv = inval;    // padded row 1: no accumulation
      a[jj] = nv;
      carry = old;
      emit<MODE>(dstb, j, t, nv, inval, l1, l2);
    }
    pb ^= 1;
    bnd[pb * kNT + tid] = a[kR - 1];
    __syncthreads();
  }

  if (MODE == 2 && tid == 0)                  // ctc_loss = bwd[:,0,0] == -inf always
    dst[(size_t)kB * kS * kT + b] = NINF;
}

// ws path: out = sanitize(out + ws - lp), plus ctc losses
__global__ void combine_kernel(const float* __restrict__ lp, const float* __restrict__ ws,
                               float* __restrict__ out) {
  const size_t N = (size_t)kB * kS * kT;
  size_t i = ((size_t)blockIdx.x * blockDim.x + threadIdx.x) * 4;
  if (i < N) {
    float4 f = *(const float4*)(out + i);
    float4 w = *(const float4*)(ws + i);
    float4 p = *(const float4*)(lp + i);
    float r[4] = {f.x + w.x - p.x, f.y + w.y - p.y, f.z + w.z - p.z, f.w + w.w - p.w};
#pragma unroll
    for (int q = 0; q < 4; ++q) {
      float v = r[q];
      if (v != v || v == NINF) v = NINF;
      else if (v == PINF)      v = 3.4028234663852886e38f;
      r[q] = v;
    }
    *(float4*)(out + i) = make_float4(r[0], r[1], r[2], r[3]);
  }
  if (blockIdx.x == 0 && threadIdx.x < kB) out[N + threadIdx.x] = NINF;
}

extern "C" void kernel_launch(void* const* d_in, const int* in_sizes, int n_in,
                              void* d_out, int out_size, void* d_ws, size_t ws_size,
                              hipStream_t stream) {
  (void)in_sizes; (void)n_in; (void)out_size;
  const float* lp  = (const float*)d_in[0];
  const int*   tls = (const int*)d_in[1];
  const int*   ils = (const int*)d_in[2];
  float* out = (float*)d_out;
  const size_t N = (size_t)kB * kS * kT;
  const size_t smem = kSmemFloats * sizeof(float);   // 51200 B dynamic LDS

  dp_kernel<0><<<kB, kNT, smem, stream>>>(lp, tls, ils, out);
  if (ws_size >= N * sizeof(float)) {
    float* ws = (float*)d_ws;
    dp_kernel<1><<<kB, kNT, smem, stream>>>(lp, tls, ils, ws);
    combine_kernel<<<(unsigned)(N / 4 / 256), 256, 0, stream>>>(lp, ws, out);
  } else {
    dp_kernel<2><<<kB, kNT, smem, stream>>>(lp, tls, ils, out);
  }
}